// TokenCL_53085795778905
// MI455X (gfx1250) — compile-verified
//
#include <hip/hip_runtime.h>
#include <math.h>

typedef float v2f __attribute__((ext_vector_type(2)));
typedef float v8f __attribute__((ext_vector_type(8)));

#define HIDDEN   1024
#define EMB      128
#define NTOK     512
#define FSTRIDE  640     // 5 feature blocks of 128: [mu, sigma, invsig, musq, mu/sig]
#define EPSF     1e-14f

// ---------------------------------------------------------------------------
// Kernel 1: projection.  One wave (32 lanes) computes a 16(M) x 16(N) tile of
// BOTH mu and z = h@W_sigma + b_sigma, sharing the A fragment.
// A fragment (fp32 16x4): lanes 0-15 hold K={k,k+1}, lanes 16-31 K={k+2,k+3}.
// ---------------------------------------------------------------------------
__global__ __launch_bounds__(32)
void tcl_proj(const float* __restrict__ X,
              const float* __restrict__ Wmu, const float* __restrict__ bmu,
              const float* __restrict__ Wsg, const float* __restrict__ bsg,
              float* __restrict__ F)
{
    const int lane = threadIdx.x;
    const int half = lane >> 4;
    const int l    = lane & 15;
    const int i0   = blockIdx.y * 16;
    const int n0   = blockIdx.x * 16;

    const float* xrow = X + (size_t)(i0 + l) * HIDDEN + 2 * half;
    const float* wm   = Wmu + n0 + l;
    const float* ws   = Wsg + n0 + l;

    v8f cm = {};
    v8f cz = {};

#pragma unroll 2
    for (int k = 0; k < HIDDEN; k += 4) {
        const int kr = k + 2 * half;
        v2f a, bm, bs;
        a.x  = fmaxf(xrow[k],     0.0f);          // relu fused into A load
        a.y  = fmaxf(xrow[k + 1], 0.0f);
        bm.x = wm[(size_t)kr * EMB];
        bm.y = wm[(size_t)(kr + 1) * EMB];
        bs.x = ws[(size_t)kr * EMB];
        bs.y = ws[(size_t)(kr + 1) * EMB];
        __builtin_prefetch(&wm[(size_t)(kr + 32) * EMB], 0, 1);   // global_prefetch_b8
        cm = __builtin_amdgcn_wmma_f32_16x16x4_f32(false, a, false, bm, (short)0, cm, false, false);
        cz = __builtin_amdgcn_wmma_f32_16x16x4_f32(false, a, false, bs, (short)0, cz, false, false);
    }

    const float bmv = bmu[n0 + l];
    const float bsv = bsg[n0 + l];
#pragma unroll
    for (int v = 0; v < 8; ++v) {
        const int r  = i0 + v + 8 * half;         // C layout: row = v + 8*(lane>=16)
        const float mu = cm[v] + bmv;
        const float z  = cz[v] + bsv;
        const float sg = (z > 0.0f) ? (z + 1.0f + EPSF) : (expf(z) + EPSF); // elu(z)+1+eps
        F[(size_t)r * FSTRIDE + n0 + l]       = mu;
        F[(size_t)r * FSTRIDE + 128 + n0 + l] = sg;
    }
}

// ---------------------------------------------------------------------------
// Kernel 2: per-row derived features + q_i = sum(mu^2/sigma); zero accumulators
// ---------------------------------------------------------------------------
__global__ __launch_bounds__(128)
void tcl_feat(float* __restrict__ F, float* __restrict__ qi,
              float* __restrict__ accN, float* __restrict__ accD, float* __restrict__ accC)
{
    const int i = blockIdx.x;
    const int t = threadIdx.x;
    float* row = F + (size_t)i * FSTRIDE;
    const float mu  = row[t];
    const float sg  = row[128 + t];
    const float inv = 1.0f / sg;
    row[256 + t] = inv;            // 1/sigma
    row[384 + t] = mu * mu;        // mu^2
    row[512 + t] = mu * inv;       // mu/sigma

    __shared__ float red[128];
    red[t] = mu * mu * inv;
    __syncthreads();
    for (int s = 64; s > 0; s >>= 1) {
        if (t < s) red[t] += red[t + s];
        __syncthreads();
    }
    if (t == 0) {
        qi[i]   = red[0];
        accN[i] = 0.0f;
        accD[i] = 0.0f;
        accC[i] = 0.0f;
    }
}

// ---------------------------------------------------------------------------
// Kernel 3: exact all_diff / valid / label masks -> byte per (i,j)
// bit0 = den_mask, bit1 = num_mask
// ---------------------------------------------------------------------------
__global__ __launch_bounds__(256)
void tcl_mask(const float* __restrict__ F,
              const int* __restrict__ labels, const int* __restrict__ vmask,
              unsigned char* __restrict__ mb)
{
    const int idx = blockIdx.x * 256 + threadIdx.x;
    const int i = idx >> 9;
    const int j = idx & 511;
    const int li = labels[i];
    const int lj = labels[j];
    const bool vi = (vmask[i] == 1) && (li > 0);
    const bool vj = (vmask[j] == 1) && (lj > 0);

    bool alldiff = true;
    const float4* pi = (const float4*)(F + (size_t)i * FSTRIDE);  // mu block
    const float4* pj = (const float4*)(F + (size_t)j * FSTRIDE);
#pragma unroll 8
    for (int d = 0; d < 32; ++d) {
        const float4 a = pi[d];
        const float4 b = pj[d];
        if (a.x == b.x || a.y == b.y || a.z == b.z || a.w == b.w) alldiff = false;
    }
    const bool den = alldiff && vi && vj;
    const bool num = den && (li == lj);
    mb[idx] = (unsigned char)((den ? 1 : 0) | (num ? 2 : 0));
}

// ---------------------------------------------------------------------------
// Kernel 4: pairwise KL as a 512x512x768 fp32 WMMA GEMM over 6 feature-block
// pairs, fused with exp/mask epilogue and per-row reductions.
//   K[i,j] = <iv_i,sg_j>+<sg_i,iv_j>+<iv_i,mq_j>+<mq_i,iv_j>-2<a_i,mu_j>-2<mu_i,a_j>
//   loss/T = -0.5*(K + q_i + q_j - 2D)
// ---------------------------------------------------------------------------
__global__ __launch_bounds__(32)
void tcl_pair(const float* __restrict__ F, const float* __restrict__ qi,
              const unsigned char* __restrict__ mb,
              float* __restrict__ accN, float* __restrict__ accD, float* __restrict__ accC)
{
    const int lane = threadIdx.x;
    const int half = lane >> 4;
    const int l    = lane & 15;
    const int i0   = blockIdx.y * 16;
    const int j0   = blockIdx.x * 16;

    const int   ua[6] = {2, 1, 2, 3, 4, 0};   // i-side feature block per term
    const int   vb[6] = {1, 2, 3, 2, 0, 4};   // j-side feature block per term
    const float sv[6] = {1.f, 1.f, 1.f, 1.f, -2.f, -2.f};

    const float* arow = F + (size_t)(i0 + l) * FSTRIDE + 2 * half;
    const float* brow = F + (size_t)(j0 + l) * FSTRIDE + 2 * half;

    v8f c = {};
    for (int b = 0; b < 6; ++b) {
        const float* ap = arow + ua[b] * 128;
        const float* bp = brow + vb[b] * 128;
        const float  s  = sv[b];
#pragma unroll 4
        for (int k = 0; k < 128; k += 4) {
            v2f af, bf;
            af.x = ap[k];
            af.y = ap[k + 1];
            bf.x = s * bp[k];
            bf.y = s * bp[k + 1];
            c = __builtin_amdgcn_wmma_f32_16x16x4_f32(false, af, false, bf, (short)0, c, false, false);
        }
    }

    const float qj = qi[j0 + l];
#pragma unroll
    for (int v = 0; v < 8; ++v) {
        const int i = i0 + v + 8 * half;
        const int j = j0 + l;
        const float lt = -0.5f * (c[v] + qi[i] + qj - 256.0f);  // loss / CL_TEMPERATURE
        const float e  = expf(lt);
        const unsigned char m = mb[i * 512 + j];
        float en = (m & 2) ? e : 0.0f;
        float ed = (m & 1) ? e : 0.0f;
        float cn = (m & 2) ? 1.0f : 0.0f;
        // butterfly reduce across the 16 lanes of each wave half
        for (int off = 8; off >= 1; off >>= 1) {
            en += __shfl_xor(en, off, 32);
            ed += __shfl_xor(ed, off, 32);
            cn += __shfl_xor(cn, off, 32);
        }
        if (l == 0) {   // lanes 0 and 16 commit rows v and v+8
            atomicAdd(&accN[i], en);
            atomicAdd(&accD[i], ed);
            atomicAdd(&accC[i], cn);
        }
    }
}

// ---------------------------------------------------------------------------
// Kernel 5: final scalar
// ---------------------------------------------------------------------------
__global__ __launch_bounds__(512)
void tcl_final(const float* __restrict__ accN, const float* __restrict__ accD,
               const float* __restrict__ accC, float* __restrict__ out)
{
    const int t = threadIdx.x;
    const float n  = accN[t];
    const float d  = accD[t];
    const float cc = accC[t];
    const bool inc = cc > 0.0f;
    const float lf = inc ? (-logf(n) + logf(d) + logf(cc)) : 0.0f;
    const float ic = inc ? 1.0f : 0.0f;

    __shared__ float r1[512];
    __shared__ float r2[512];
    r1[t] = lf;
    r2[t] = ic;
    __syncthreads();
    for (int s = 256; s > 0; s >>= 1) {
        if (t < s) { r1[t] += r1[t + s]; r2[t] += r2[t + s]; }
        __syncthreads();
    }
    if (t == 0) out[0] = r1[0] / fmaxf(r2[0], 1.0f);
}

// ---------------------------------------------------------------------------
extern "C" void kernel_launch(void* const* d_in, const int* in_sizes, int n_in,
                              void* d_out, int out_size, void* d_ws, size_t ws_size,
                              hipStream_t stream)
{
    const float* X    = (const float*)d_in[0];   // [512,1024]
    const int*   lab  = (const int*)  d_in[1];   // [512]
    const int*   msk  = (const int*)  d_in[2];   // [512]
    const float* Wmu  = (const float*)d_in[3];   // [1024,128]
    const float* bmu  = (const float*)d_in[4];   // [128]
    const float* Wsg  = (const float*)d_in[5];   // [1024,128]
    const float* bsg  = (const float*)d_in[6];   // [128]
    float*       out  = (float*)d_out;

    // workspace layout
    float* F    = (float*)d_ws;                     // 512*640 floats
    float* qi   = F + (size_t)NTOK * FSTRIDE;       // 512
    float* accN = qi + NTOK;                        // 512
    float* accD = accN + NTOK;                      // 512
    float* accC = accD + NTOK;                      // 512
    unsigned char* mb = (unsigned char*)(accC + NTOK);  // 512*512 bytes
    (void)in_sizes; (void)n_in; (void)out_size; (void)ws_size;

    tcl_proj <<<dim3(EMB / 16, NTOK / 16), 32, 0, stream>>>(X, Wmu, bmu, Wsg, bsg, F);
    tcl_feat <<<NTOK, 128, 0, stream>>>(F, qi, accN, accD, accC);
    tcl_mask <<<(NTOK * NTOK) / 256, 256, 0, stream>>>(F, lab, msk, mb);
    tcl_pair <<<dim3(NTOK / 16, NTOK / 16), 32, 0, stream>>>(F, qi, mb, accN, accD, accC);
    tcl_final<<<1, 512, 0, stream>>>(accN, accD, accC, out);
}